// JointEncoder_72507637891628
// MI455X (gfx1250) — compile-verified
//
#include <hip/hip_runtime.h>

// Shapes from the reference
#define B_   8
#define N_   20
#define T_   300
#define U_   64
#define H_   512
#define NT_  6000          // N_*T_
#define EPSF 1e-13f

typedef float v2f __attribute__((ext_vector_type(2)));
typedef float v8f __attribute__((ext_vector_type(8)));

// CDNA5 f32 WMMA: D(16x16) = A(16x4) * B(4x16) + C.
// Layouts (ISA 7.12.2): A lane l<16: M=l, vgpr j -> K=j ; lane>=16: K=2+j.
//                       B lane l<16: row K=j ; lane>=16: row K=2+j ; N=l%16.
//                       C/D vgpr i, lane l: M = i + 8*(l>=16), N = l%16.
__device__ __forceinline__ v8f wmma4(v2f a, v2f b, v8f c) {
  return __builtin_amdgcn_wmma_f32_16x16x4_f32(false, a, false, b, (short)0, c,
                                               false, false);
}

// ---------------------------------------------------------------------------
// K1: scores[b, k, u] = sum_h doc[b,k,h] * q[b,u,h]   (M=6000, N=64, K=512)
// q[b] staged in LDS with pitch 516 -> conflict-free transposed B reads.
// ---------------------------------------------------------------------------
__global__ __launch_bounds__(256) void k_scores(const float* __restrict__ doc,
                                                const float* __restrict__ q,
                                                float* __restrict__ scores) {
  __shared__ __align__(16) float sq[U_ * 516];   // 132 KB (<320 KB WGP LDS)
  const int b = blockIdx.y;
  const float* qb = q + (size_t)b * U_ * H_;
  for (int i4 = threadIdx.x; i4 < U_ * H_ / 4; i4 += 256) {
    int u  = i4 >> 7;               // / (H_/4)
    int k4 = (i4 & 127) << 2;
    *(float4*)&sq[u * 516 + k4] = ((const float4*)qb)[i4];
  }
  __syncthreads();

  const int wave = threadIdx.x >> 5;
  const int lane = threadIdx.x & 31;
  const int lrow = lane & 15, lhi = lane >> 4;
  const int mtile = blockIdx.x * 8 + wave;       // 16-row strip per wave
  if (mtile * 16 >= NT_) return;                 // wave-uniform

  const float* arow = doc + ((size_t)b * NT_ + mtile * 16 + lrow) * H_;
  v8f acc0 = {}, acc1 = {}, acc2 = {}, acc3 = {};
#pragma unroll 2
  for (int k = 0; k < H_; k += 4) {
    v2f a = *(const v2f*)&arow[k + 2 * lhi];
    v2f b0 = *(const v2f*)&sq[(0  + lrow) * 516 + k + 2 * lhi];
    v2f b1 = *(const v2f*)&sq[(16 + lrow) * 516 + k + 2 * lhi];
    v2f b2 = *(const v2f*)&sq[(32 + lrow) * 516 + k + 2 * lhi];
    v2f b3 = *(const v2f*)&sq[(48 + lrow) * 516 + k + 2 * lhi];
    acc0 = wmma4(a, b0, acc0);
    acc1 = wmma4(a, b1, acc1);
    acc2 = wmma4(a, b2, acc2);
    acc3 = wmma4(a, b3, acc3);
  }
  float* srow = scores + ((size_t)b * NT_ + mtile * 16) * U_;
#pragma unroll
  for (int i = 0; i < 8; ++i) {
    const size_t r = (size_t)(i + 8 * lhi) * U_;
    srow[r + 0  + lrow] = acc0[i];
    srow[r + 16 + lrow] = acc1[i];
    srow[r + 32 + lrow] = acc2[i];
    srow[r + 48 + lrow] = acc3[i];
  }
}

// ---------------------------------------------------------------------------
// K2: attn = masked_softmax(scores, qm) along U.
// final_u = e_u*m_u / (Sum(e*m) + eps*Sum(e)), e_u = exp(s_u*m_u - max)
// ---------------------------------------------------------------------------
__global__ __launch_bounds__(256) void k_softmax_u(const float* __restrict__ scores,
                                                   const int* __restrict__ q_mask,
                                                   float* __restrict__ attn) {
  __shared__ float tile[64 * 65];
  __shared__ float qm[U_];
  __shared__ float rinv[64];
  const int b  = blockIdx.y;
  const int r0 = blockIdx.x * 64;
  const int rows = (NT_ - r0 < 64) ? (NT_ - r0) : 64;
  if (threadIdx.x < U_) qm[threadIdx.x] = (float)q_mask[b * U_ + threadIdx.x];
  for (int idx = threadIdx.x; idx < rows * 64; idx += 256) {
    int r = idx >> 6, u = idx & 63;
    tile[r * 65 + u] = scores[((size_t)b * NT_ + r0 + r) * U_ + u];
  }
  __syncthreads();
  if (threadIdx.x < rows) {
    const int r = threadIdx.x;
    float m = -3.402823466e+38f;
    for (int u = 0; u < U_; ++u) m = fmaxf(m, tile[r * 65 + u] * qm[u]);
    float Z = 0.f, Sm = 0.f;
    for (int u = 0; u < U_; ++u) {
      float e  = __expf(tile[r * 65 + u] * qm[u] - m);
      float em = e * qm[u];
      tile[r * 65 + u] = em;
      Z += e; Sm += em;
    }
    rinv[r] = 1.0f / (Sm + EPSF * Z);
  }
  __syncthreads();
  for (int idx = threadIdx.x; idx < rows * 64; idx += 256) {
    int r = idx >> 6, u = idx & 63;
    attn[((size_t)b * NT_ + r0 + r) * U_ + u] = tile[r * 65 + u] * rinv[r];
  }
}

// ---------------------------------------------------------------------------
// K3: attn_t[b,u, n*T+t] = masked_softmax over t of scores[b, n*T+t, u]*dm.
// One block per (b,n): 300x64 tile in padded LDS.
// ---------------------------------------------------------------------------
__global__ __launch_bounds__(256) void k_softmax_t(const float* __restrict__ scores,
                                                   const int* __restrict__ doc_mask,
                                                   float* __restrict__ attn_t) {
  __shared__ float tile[T_ * 65];                // 78 KB
  __shared__ float dm[T_];
  __shared__ float rinv[U_];
  const int b = blockIdx.y, n = blockIdx.x;
  for (int t = threadIdx.x; t < T_; t += 256)
    dm[t] = (float)doc_mask[((size_t)b * N_ + n) * T_ + t];
  for (int idx = threadIdx.x; idx < T_ * U_; idx += 256) {
    int t = idx >> 6, u = idx & 63;
    tile[t * 65 + u] = scores[((size_t)b * NT_ + n * T_ + t) * U_ + u];
  }
  __syncthreads();
  if (threadIdx.x < U_) {
    const int u = threadIdx.x;
    float m = -3.402823466e+38f;
    for (int t = 0; t < T_; ++t) m = fmaxf(m, tile[t * 65 + u] * dm[t]);
    float Z = 0.f, Sm = 0.f;
    for (int t = 0; t < T_; ++t) {
      float e  = __expf(tile[t * 65 + u] * dm[t] - m);
      float em = e * dm[t];
      tile[t * 65 + u] = em;
      Z += e; Sm += em;
    }
    rinv[u] = 1.0f / (Sm + EPSF * Z);
  }
  __syncthreads();
  for (int idx = threadIdx.x; idx < U_ * T_; idx += 256) {
    int u = idx / T_, t = idx - u * T_;
    attn_t[((size_t)b * U_ + u) * NT_ + n * T_ + t] = tile[t * 65 + u] * rinv[u];
  }
}

// ---------------------------------------------------------------------------
// K4: aggdoc[b] = attn_t(64 x 6000) @ doc(6000 x 512).  Wave: 16x32 tile.
// ---------------------------------------------------------------------------
__global__ __launch_bounds__(256) void k_aggdoc(const float* __restrict__ attn_t,
                                                const float* __restrict__ doc,
                                                float* __restrict__ aggdoc) {
  const int b    = blockIdx.y;
  const int wave = threadIdx.x >> 5, lane = threadIdx.x & 31;
  const int lrow = lane & 15, lhi = lane >> 4;
  const int row0 = (wave & 3) * 16;
  const int col0 = blockIdx.x * 64 + (wave >> 2) * 32;

  const float* arow = attn_t + ((size_t)b * U_ + row0 + lrow) * NT_;
  const float* bb   = doc + (size_t)b * NT_ * H_;
  v8f acc0 = {}, acc1 = {};
#pragma unroll 2
  for (int k = 0; k < NT_; k += 4) {
    v2f a = *(const v2f*)&arow[k + 2 * lhi];
    const float* r0p = bb + (size_t)(k + 2 * lhi) * H_;
    const float* r1p = r0p + H_;
    v2f b0; b0.x = r0p[col0 + lrow];      b0.y = r1p[col0 + lrow];
    v2f b1; b1.x = r0p[col0 + 16 + lrow]; b1.y = r1p[col0 + 16 + lrow];
    acc0 = wmma4(a, b0, acc0);
    acc1 = wmma4(a, b1, acc1);
  }
  float* dst = aggdoc + ((size_t)b * U_ + row0) * H_;
#pragma unroll
  for (int i = 0; i < 8; ++i) {
    const size_t r = (size_t)(i + 8 * lhi) * H_;
    dst[r + col0 + lrow]      = acc0[i];
    dst[r + col0 + 16 + lrow] = acc1[i];
  }
}

// ---------------------------------------------------------------------------
// K5: fused epilogue.  out[:, 0:512]  = attn @ q        (aggq)
//                      out[:, 512:1024] = attn @ aggdoc (aggq2)
// Both 64x512 B matrices live in LDS (256 KB); attn A-frags loaded once per
// wave and reused across all 64 output n-tiles.
// ---------------------------------------------------------------------------
__global__ __launch_bounds__(256) void k_output(const float* __restrict__ attn,
                                                const float* __restrict__ q,
                                                const float* __restrict__ aggdoc,
                                                float* __restrict__ out) {
  __shared__ __align__(16) float sb[2 * U_ * H_];   // 256 KB
  const int b = blockIdx.y;
  {
    const float4* qs = (const float4*)(q + (size_t)b * U_ * H_);
    const float4* as = (const float4*)(aggdoc + (size_t)b * U_ * H_);
    float4* d0 = (float4*)sb;
    float4* d1 = (float4*)(sb + U_ * H_);
    for (int i = threadIdx.x; i < U_ * H_ / 4; i += 256) { d0[i] = qs[i]; d1[i] = as[i]; }
  }
  __syncthreads();

  const int wave = threadIdx.x >> 5, lane = threadIdx.x & 31;
  const int lrow = lane & 15, lhi = lane >> 4;
  const int mtile = blockIdx.x * 8 + wave;
  if (mtile * 16 >= NT_) return;                    // wave-uniform

  v2f af[16];                                       // full 16x64 A tile
  const float* arow = attn + ((size_t)b * NT_ + mtile * 16 + lrow) * U_;
#pragma unroll
  for (int kk = 0; kk < 16; ++kk) af[kk] = *(const v2f*)&arow[kk * 4 + 2 * lhi];

  float* orow = out + ((size_t)b * NT_ + mtile * 16) * (2 * H_);
#pragma unroll
  for (int half = 0; half < 2; ++half) {
    const float* bl = sb + half * U_ * H_;
    for (int nt = 0; nt < 32; ++nt) {
      const int n0 = nt * 16;
      v8f acc = {};
#pragma unroll
      for (int kk = 0; kk < 16; ++kk) {
        const int kb = kk * 4 + 2 * lhi;
        v2f bb; bb.x = bl[kb * H_ + n0 + lrow]; bb.y = bl[(kb + 1) * H_ + n0 + lrow];
        acc = wmma4(af[kk], bb, acc);
      }
#pragma unroll
      for (int i = 0; i < 8; ++i)
        orow[(size_t)(i + 8 * lhi) * (2 * H_) + half * H_ + n0 + lrow] = acc[i];
    }
  }
}

// ---------------------------------------------------------------------------
extern "C" void kernel_launch(void* const* d_in, const int* in_sizes, int n_in,
                              void* d_out, int out_size, void* d_ws, size_t ws_size,
                              hipStream_t stream) {
  const float* doc      = (const float*)d_in[0];
  const float* q        = (const float*)d_in[1];
  const int*   doc_mask = (const int*)d_in[2];
  const int*   q_mask   = (const int*)d_in[3];

  float* ws     = (float*)d_ws;
  float* scores = ws;                                   // B*NT*U
  float* attn   = scores + (size_t)B_ * NT_ * U_;       // B*NT*U
  float* attn_t = attn   + (size_t)B_ * NT_ * U_;       // B*U*NT
  float* aggd   = attn_t + (size_t)B_ * NT_ * U_;       // B*U*H
  float* outp   = (float*)d_out;

  const dim3 blk(256);
  const int mblocks = (NT_ / 16 + 7) / 8;               // 47

  k_scores   <<<dim3(mblocks, B_), blk, 0, stream>>>(doc, q, scores);
  k_softmax_u<<<dim3((NT_ + 63) / 64, B_), blk, 0, stream>>>(scores, q_mask, attn);
  k_softmax_t<<<dim3(N_, B_), blk, 0, stream>>>(scores, doc_mask, attn_t);
  k_aggdoc   <<<dim3(H_ / 64, B_), blk, 0, stream>>>(attn_t, doc, aggd);
  k_output   <<<dim3(mblocks, B_), blk, 0, stream>>>(attn, q, aggd, outp);
}